// PIXBAModel_58583353917861
// MI455X (gfx1250) — compile-verified
//
#include <hip/hip_runtime.h>
#include <hip/hip_bf16.h>
#include <math.h>

// ---------------- model dims ----------------
#define B_       8
#define IMG_     224
#define PATCH_   16
#define C_IN_    3
#define GRID_    14
#define NPATCH_  196
#define DM_      768
#define NLAYERS_ 12
#define DSTATE_  16
#define DCONV_   4
#define DINNER_  1536
#define DTRANK_  48
#define LKEEP_   147
#define LTOK_    148                 // 1 cls + 147 kept patches
#define MTOK_    (B_ * LTOK_)        // 1184 (= 37 * 32)
#define XDBL_N_  (DTRANK_ + 2 * DSTATE_)   // 80

typedef float v2f __attribute__((ext_vector_type(2)));
typedef float v8f __attribute__((ext_vector_type(8)));

// ---------------- helpers ----------------
__device__ __forceinline__ float silu_f(float x) {
    return x / (1.f + __expf(-x));
}
__device__ __forceinline__ float softplus_f(float x) {
    return (x > 20.f) ? x : log1pf(__expf(x));
}

// ---------------- im2col for patch embed ----------------
// A[b*196+p][c*256+py*16+px] = pixels[b][c][gy*16+py][gx*16+px]
__global__ void pixba_im2col(const float* __restrict__ px, float* __restrict__ A) {
    int row = blockIdx.x;              // b*196 + p
    int b = row / NPATCH_;
    int p = row % NPATCH_;
    int gy = p / GRID_, gx = p % GRID_;
    for (int col = threadIdx.x; col < DM_; col += blockDim.x) {
        int c   = col >> 8;
        int py  = (col >> 4) & 15;
        int pxi = col & 15;
        A[(size_t)row * DM_ + col] =
            px[(((size_t)b * C_IN_ + c) * IMG_ + (gy * PATCH_ + py)) * IMG_ + gx * PATCH_ + pxi];
    }
}

// ---------------- WMMA fp32 GEMM: C[M][N] = A[M][K] * W[N][K]^T ----------------
// One wave computes a (16*MREP) x (16*NREP) tile. K and LDA are compile-time so
// every fragment address is base + immediate offset -> the whole k-step's loads
// can be clause-batched ahead of the MREP*NREP WMMAs (which also form that many
// independent accumulator chains, hiding the f32 WMMA RAW latency).
// A-frag layout (ISA 7.12.2, 32-bit A 16x4): lane<16 -> K={0,1}, lane>=16 -> K={2,3}, M=lane&15.
// B-frag mirrored: v0 rows K={0,2}, v1 rows K={1,3}, N=lane&15.
// C/D layout: vgpr j -> row (lane>>4)*8 + j, col lane&15.
// EPI: 0 = (+bias) store; 1 = +bias +pos_embed[(row%196)+1]; 2 = softplus(+bias); 3 = C += acc
template <int EPI, int MREP, int NREP, int K, int LDA>
__global__ void pixba_gemm_wmma(const float* __restrict__ A,
                                const float* __restrict__ W,
                                float* __restrict__ C, int ldc,
                                const float* __restrict__ bias,
                                const float* __restrict__ bias2) {
    const int lane = threadIdx.x & 31;
    const int m0 = blockIdx.y * (16 * MREP);
    const int n0 = blockIdx.x * (16 * NREP);
    const int r  = lane & 15;
    const int kk = (lane >> 4) << 1;   // 0 or 2

    const float* __restrict__ arow = A + (size_t)(m0 + r) * LDA + kk;
    const float* __restrict__ wrow = W + (size_t)(n0 + r) * K + kk;

    v8f acc[MREP][NREP];
#pragma unroll
    for (int m = 0; m < MREP; ++m)
#pragma unroll
        for (int i = 0; i < NREP; ++i) acc[m][i] = (v8f){};

    for (int k0 = 0; k0 < K; k0 += 16) {
        // CDNA5 prefetch (global_prefetch_b8): pull next chunks toward near caches
        __builtin_prefetch(arow + k0 + 64, 0, 1);
        __builtin_prefetch(wrow + k0 + 64, 0, 1);
#pragma unroll
        for (int k = 0; k < 16; k += 4) {
            // load ALL fragments of this k-step first (clause-batched loads)
            v2f a[MREP], bm[NREP];
#pragma unroll
            for (int m = 0; m < MREP; ++m) {
                a[m].x = arow[m * 16 * LDA + k0 + k];
                a[m].y = arow[m * 16 * LDA + k0 + k + 1];
            }
#pragma unroll
            for (int i = 0; i < NREP; ++i) {
                bm[i].x = wrow[i * 16 * K + k0 + k];
                bm[i].y = wrow[i * 16 * K + k0 + k + 1];
            }
#pragma unroll
            for (int m = 0; m < MREP; ++m)
#pragma unroll
                for (int i = 0; i < NREP; ++i)
                    acc[m][i] = __builtin_amdgcn_wmma_f32_16x16x4_f32(
                        /*neg_a=*/false, a[m], /*neg_b=*/false, bm[i],
                        /*c_mod=*/(short)0, acc[m][i],
                        /*reuse_a=*/false, /*reuse_b=*/false);
        }
    }

#pragma unroll
    for (int m = 0; m < MREP; ++m) {
        const int rbase = m0 + m * 16 + ((lane >> 4) << 3);
#pragma unroll
        for (int i = 0; i < NREP; ++i) {
            const int col = n0 + i * 16 + r;
#pragma unroll
            for (int j = 0; j < 8; ++j) {
                const int row = rbase + j;
                float v = acc[m][i][j];
                if (EPI == 0) {
                    if (bias) v += bias[col];
                    C[(size_t)row * ldc + col] = v;
                } else if (EPI == 1) {
                    v += bias[col] + bias2[(size_t)((row % NPATCH_) + 1) * DM_ + col];
                    C[(size_t)row * ldc + col] = v;
                } else if (EPI == 2) {
                    C[(size_t)row * ldc + col] = softplus_f(v + bias[col]);
                } else {
                    C[(size_t)row * ldc + col] += v;
                }
            }
        }
    }
}

// ---------------- random masking: O(n^2) rank over 196 values/row ----------------
__global__ void pixba_mask(const float* __restrict__ noise, int* __restrict__ ids_shuffle,
                           float* __restrict__ out_mask, float* __restrict__ out_idsr) {
    int b = blockIdx.x;
    int i = threadIdx.x;
    if (i >= NPATCH_) return;
    const float* nr = noise + (size_t)b * NPATCH_;
    float ni = nr[i];
    int rank = 0;
    for (int j = 0; j < NPATCH_; ++j) {
        float nj = nr[j];
        rank += (nj < ni) || (nj == ni && j < i);
    }
    ids_shuffle[b * NPATCH_ + rank] = i;             // argsort scatter
    out_mask[b * NPATCH_ + i] = (rank < LKEEP_) ? 0.f : 1.f;
    out_idsr[b * NPATCH_ + i] = (float)rank;         // ids_restore
}

// ---------------- gather kept tokens + prepend cls ----------------
__global__ void pixba_build_h(const float* __restrict__ patch_out, const int* __restrict__ ids_shuffle,
                              const float* __restrict__ cls_token, const float* __restrict__ pos_embed,
                              float* __restrict__ h) {
    int t = blockIdx.x % LTOK_;
    int b = blockIdx.x / LTOK_;
    float* dst = h + ((size_t)b * LTOK_ + t) * DM_;
    if (t == 0) {
        for (int c = threadIdx.x; c < DM_; c += blockDim.x)
            dst[c] = cls_token[c] + pos_embed[c];
    } else {
        int p = ids_shuffle[b * NPATCH_ + (t - 1)];
        const float* src = patch_out + ((size_t)b * NPATCH_ + p) * DM_;
        for (int c = threadIdx.x; c < DM_; c += blockDim.x)
            dst[c] = src[c];
    }
}

// ---------------- layernorm (one block / token, blockDim must be 256) ----------------
__global__ void pixba_ln(const float* __restrict__ x, const float* __restrict__ w,
                         const float* __restrict__ bb, float* __restrict__ y) {
    __shared__ float red[256];
    const float* xr = x + (size_t)blockIdx.x * DM_;
    float s = 0.f;
    for (int c = threadIdx.x; c < DM_; c += blockDim.x) s += xr[c];
    red[threadIdx.x] = s; __syncthreads();
    for (int o = 128; o > 0; o >>= 1) { if (threadIdx.x < o) red[threadIdx.x] += red[threadIdx.x + o]; __syncthreads(); }
    float mu = red[0] * (1.f / DM_);
    __syncthreads();
    float v = 0.f;
    for (int c = threadIdx.x; c < DM_; c += blockDim.x) { float d = xr[c] - mu; v += d * d; }
    red[threadIdx.x] = v; __syncthreads();
    for (int o = 128; o > 0; o >>= 1) { if (threadIdx.x < o) red[threadIdx.x] += red[threadIdx.x + o]; __syncthreads(); }
    float rstd = rsqrtf(red[0] * (1.f / DM_) + 1e-12f);
    float* yr = y + (size_t)blockIdx.x * DM_;
    for (int c = threadIdx.x; c < DM_; c += blockDim.x)
        yr[c] = (xr[c] - mu) * rstd * w[c] + bb[c];
}

// ---------------- causal depthwise conv1d (k=4) + SiLU ----------------
// xs lives in xz[..., 0:1536] with row stride 3072
__global__ void pixba_conv_silu(const float* __restrict__ xz, const float* __restrict__ cw,
                                const float* __restrict__ cb, float* __restrict__ xc) {
    int idx = blockIdx.x * blockDim.x + threadIdx.x;
    if (idx >= MTOK_ * DINNER_) return;
    int d = idx % DINNER_;
    int t = idx / DINNER_;
    int l = t % LTOK_;
    int b = t / LTOK_;
    float acc = cb[d];
#pragma unroll
    for (int k = 0; k < DCONV_; ++k) {
        int ll = l - (DCONV_ - 1) + k;
        if (ll >= 0)
            acc += cw[d * DCONV_ + k] * xz[((size_t)(b * LTOK_ + ll)) * (2 * DINNER_) + d];
    }
    xc[idx] = silu_f(acc);
}

// ---------------- selective scan: one thread per (b, channel), 16 states in regs ----------------
__global__ void pixba_scan(const float* __restrict__ dtv, const float* __restrict__ xc,
                           const float* __restrict__ xdbl, const float* __restrict__ xz,
                           const float* __restrict__ A_log, const float* __restrict__ Dp,
                           float* __restrict__ ys) {
    int idx = blockIdx.x * blockDim.x + threadIdx.x;
    if (idx >= B_ * DINNER_) return;
    int d = idx % DINNER_;
    int b = idx / DINNER_;

    float Arow[DSTATE_], hstate[DSTATE_];
#pragma unroll
    for (int n = 0; n < DSTATE_; ++n) { Arow[n] = -__expf(A_log[d * DSTATE_ + n]); hstate[n] = 0.f; }
    float Dd = Dp[d];

    for (int l = 0; l < LTOK_; ++l) {
        size_t t = (size_t)b * LTOK_ + l;
        float dt = dtv[t * DINNER_ + d];
        float x  = xc[t * DINNER_ + d];
        float z  = xz[t * (2 * DINNER_) + DINNER_ + d];
        const float* Bm = xdbl + t * XDBL_N_ + DTRANK_;
        const float* Cm = Bm + DSTATE_;
        float dtx = dt * x;
        float y = 0.f;
#pragma unroll
        for (int n = 0; n < DSTATE_; ++n) {
            hstate[n] = __expf(dt * Arow[n]) * hstate[n] + dtx * Bm[n];
            y += hstate[n] * Cm[n];
        }
        y += x * Dd;
        y *= silu_f(z);
        ys[t * DINNER_ + d] = y;
    }
}

// ---------------- driver ----------------
extern "C" void kernel_launch(void* const* d_in, const int* in_sizes, int n_in,
                              void* d_out, int out_size, void* d_ws, size_t ws_size,
                              hipStream_t stream) {
    (void)in_sizes; (void)n_in; (void)out_size; (void)ws_size;
    const float* pixel_values = (const float*)d_in[0];
    const float* noise        = (const float*)d_in[1];
    const float* conv_w       = (const float*)d_in[2];
    const float* conv_b       = (const float*)d_in[3];
    const float* cls_token    = (const float*)d_in[4];
    const float* pos_embed    = (const float*)d_in[5];
    const float* ln_w         = (const float*)d_in[6];
    const float* ln_b         = (const float*)d_in[7];
    const float* W_in         = (const float*)d_in[8];
    const float* conv1d_w     = (const float*)d_in[9];
    const float* conv1d_b     = (const float*)d_in[10];
    const float* W_x          = (const float*)d_in[11];
    const float* W_dt         = (const float*)d_in[12];
    const float* dt_bias      = (const float*)d_in[13];
    const float* A_log        = (const float*)d_in[14];
    const float* Dp           = (const float*)d_in[15];
    const float* W_out        = (const float*)d_in[16];
    const float* lnf_w        = (const float*)d_in[17];
    const float* lnf_b        = (const float*)d_in[18];

    float* ws = (float*)d_ws;
    // workspace layout (floats); buf0/buf1 are reused across phases
    const size_t SZ_IC   = (size_t)B_ * NPATCH_ * DM_;          // 1,204,224
    float* A_ic      = ws;                                      // buf0: im2col, later h
    float* hbuf      = ws;                                      //  (h fits inside buf0)
    float* patch_out = ws + SZ_IC;                              // buf1: patch embed, later xn
    float* xn        = ws + SZ_IC;
    float* xz        = ws + 2 * SZ_IC;                                   // 1184*3072
    float* xc        = xz + (size_t)MTOK_ * 2 * DINNER_;                 // 1184*1536
    float* xdbl      = xc + (size_t)MTOK_ * DINNER_;                     // 1184*80
    float* dtv       = xdbl + (size_t)MTOK_ * XDBL_N_;                   // 1184*1536
    float* ysb       = dtv + (size_t)MTOK_ * DINNER_;                    // 1184*1536
    int*   ids_shuf  = (int*)(ysb + (size_t)MTOK_ * DINNER_);            // 8*196 ints

    float* out      = (float*)d_out;
    float* out_h    = out;                                      // 8*148*768
    float* out_mask = out + (size_t)MTOK_ * DM_;                // 8*196
    float* out_idsr = out_mask + (size_t)B_ * NPATCH_;          // 8*196

    // 1) patch embed as WMMA GEMM: im2col -> (1568x768) x (768x768)^T + conv_b + pos_embed[1:]
    pixba_im2col<<<B_ * NPATCH_, 256, 0, stream>>>(pixel_values, A_ic);
    pixba_gemm_wmma<1, 2, 4, DM_, DM_>
        <<<dim3(DM_ / 64, (B_ * NPATCH_) / 32), 32, 0, stream>>>(
        A_ic, conv_w, patch_out, DM_, conv_b, pos_embed);

    // 2) masking + outputs mask / ids_restore
    pixba_mask<<<B_, 224, 0, stream>>>(noise, ids_shuf, out_mask, out_idsr);

    // 3) gather kept tokens, prepend cls (+pos_embed[0])
    pixba_build_h<<<B_ * LTOK_, 256, 0, stream>>>(patch_out, ids_shuf, cls_token, pos_embed, hbuf);

    // 4) 12 Mamba layers
    for (int layer = 0; layer < NLAYERS_; ++layer) {
        const float* lWi = W_in     + (size_t)layer * 2 * DINNER_ * DM_;
        const float* lCw = conv1d_w + (size_t)layer * DINNER_ * DCONV_;
        const float* lCb = conv1d_b + (size_t)layer * DINNER_;
        const float* lWx = W_x      + (size_t)layer * XDBL_N_ * DINNER_;
        const float* lWd = W_dt     + (size_t)layer * DINNER_ * DTRANK_;
        const float* lDb = dt_bias  + (size_t)layer * DINNER_;
        const float* lAl = A_log    + (size_t)layer * DINNER_ * DSTATE_;
        const float* lDp = Dp       + (size_t)layer * DINNER_;
        const float* lWo = W_out    + (size_t)layer * DM_ * DINNER_;

        pixba_ln<<<MTOK_, 256, 0, stream>>>(hbuf, ln_w + layer * DM_, ln_b + layer * DM_, xn);
        // xz = xn @ W_in^T   (1184 x 768 x 3072)
        pixba_gemm_wmma<0, 2, 4, DM_, DM_>
            <<<dim3((2 * DINNER_) / 64, MTOK_ / 32), 32, 0, stream>>>(
            xn, lWi, xz, 2 * DINNER_, nullptr, nullptr);
        // depthwise causal conv + SiLU on xs half
        pixba_conv_silu<<<(MTOK_ * DINNER_ + 255) / 256, 256, 0, stream>>>(xz, lCw, lCb, xc);
        // x_dbl = xc @ W_x^T  (1184 x 1536 x 80)  -- small N: NREP=1 keeps parallelism
        pixba_gemm_wmma<0, 2, 1, DINNER_, DINNER_>
            <<<dim3(XDBL_N_ / 16, MTOK_ / 32), 32, 0, stream>>>(
            xc, lWx, xdbl, XDBL_N_, nullptr, nullptr);
        // dt = softplus(x_dbl[:, :48] @ W_dt^T + dt_bias)  (1184 x 48 x 1536), A stride 80
        pixba_gemm_wmma<2, 2, 4, DTRANK_, XDBL_N_>
            <<<dim3(DINNER_ / 64, MTOK_ / 32), 32, 0, stream>>>(
            xdbl, lWd, dtv, DINNER_, lDb, nullptr);
        // selective scan (+ D skip + SiLU(z) gate)
        pixba_scan<<<(B_ * DINNER_ + 255) / 256, 256, 0, stream>>>(
            dtv, xc, xdbl, xz, lAl, lDp, ysb);
        // h += ys @ W_out^T   (1184 x 1536 x 768)
        pixba_gemm_wmma<3, 2, 4, DINNER_, DINNER_>
            <<<dim3(DM_ / 64, MTOK_ / 32), 32, 0, stream>>>(
            ysb, lWo, hbuf, DM_, nullptr, nullptr);
    }

    // 5) final layernorm -> output h
    pixba_ln<<<MTOK_, 256, 0, stream>>>(hbuf, lnf_w, lnf_b, out_h);
}